// HardCL_75144747811041
// MI455X (gfx1250) — compile-verified
//
#include <hip/hip_runtime.h>

typedef float v2f __attribute__((ext_vector_type(2)));
typedef float v4f __attribute__((ext_vector_type(4)));
typedef float v8f __attribute__((ext_vector_type(8)));

#define B_      4096            // BATCH
#define D_      128             // DIM
#define N2_     (2 * B_)        // 8192 rows of concat matrix
#define NTILES  (N2_ / 16)      // 512 column tiles
#define KSTEPS  (D_ / 4)        // 32 wmma k-steps (K=4 each)
#define CS      4               // column splits (partial sums)
#define TPW     (NTILES / CS)   // 128 column tiles per workgroup
#define LSTRIDE 132             // padded LDS row stride (floats): ds_load_b64 conflict-free

#define TAU_PLUS   0.1f
#define INV_TEMP   2.0f         // 1 / TEMPERATURE (0.5)

__device__ __forceinline__ const float* row_ptr(const float* __restrict__ o1,
                                                const float* __restrict__ o2,
                                                int row) {
  return (row < B_) ? (o1 + (size_t)row * D_) : (o2 + (size_t)(row - B_) * D_);
}

__device__ __forceinline__ v8f wmma_f32(v2f a, v2f b, v8f c) {
  // V_WMMA_F32_16X16X4_F32: D = A(16x4) * B(4x16) + C  (fp32, exact)
  return __builtin_amdgcn_wmma_f32_16x16x4_f32(
      /*neg_a=*/false, a, /*neg_b=*/false, b,
      /*c_mod=*/(short)0, c, /*reuse_a=*/false, /*reuse_b=*/false);
}

// pos[i] = exp(dot(out_1[i], out_2[i]) / T), one wave32 per row
__global__ __launch_bounds__(256)
void hardcl_pos(const float* __restrict__ o1, const float* __restrict__ o2,
                float* __restrict__ pos) {
  const int lane = threadIdx.x & 31;
  const int row  = (blockIdx.x * blockDim.x + threadIdx.x) >> 5;
  if (row >= B_) return;
  const float* p1 = o1 + (size_t)row * D_;
  const float* p2 = o2 + (size_t)row * D_;
  float s = 0.0f;
#pragma unroll
  for (int k = lane; k < D_; k += 32) s += p1[k] * p2[k];
#pragma unroll
  for (int m = 16; m > 0; m >>= 1) s += __shfl_xor(s, m, 32);
  if (lane == 0) pos[row] = __expf(s * INV_TEMP);
}

// Each WG: 128 rows (8 waves x 16-row WMMA tiles, A resident in VGPRs),
// marches over TPW column tiles staged through double-buffered LDS.
// Writes per-column-split partial sums s1p/s2p[blockIdx.y][row].
__global__ __launch_bounds__(256)
void hardcl_rowsums(const float* __restrict__ o1, const float* __restrict__ o2,
                    float* __restrict__ s1p, float* __restrict__ s2p) {
  __shared__ float lds[2][16 * LSTRIDE];

  const int tid  = threadIdx.x;
  const int lane = tid & 31;
  const int wv   = tid >> 5;                       // 0..7
  const int row0 = blockIdx.x * 128 + wv * 16;     // this wave's 16-row tile
  const int lr   = lane & 15;                      // tile row/col index for this lane
  const int koff = (lane >> 4) << 1;               // high half-wave holds K+2,K+3
  const int t0   = blockIdx.y * TPW;               // first column tile

  // Resident A tile (16x128) in WMMA layout: contiguous float2 per lane per k-step
  const float* __restrict__ pa = row_ptr(o1, o2, row0) + lr * D_ + koff;
  v2f a[KSTEPS];
#pragma unroll
  for (int k = 0; k < KSTEPS; ++k) a[k] = *(const v2f*)(pa + 4 * k);

  // Cooperative staging indices: 256 threads x 8 floats = one 16x128 tile
  const int srow = tid >> 4;                       // 0..15
  const int scol = (tid & 15) * 8;                 // 0,8,...,120

  // Prologue: stage tile t0 into buffer 0
  {
    const float* ps = row_ptr(o1, o2, t0 * 16) + srow * D_ + scol;
    *(v4f*)(&lds[0][srow * LSTRIDE + scol])     = *(const v4f*)(ps);
    *(v4f*)(&lds[0][srow * LSTRIDE + scol + 4]) = *(const v4f*)(ps + 4);
  }

  v8f s1 = {0.f,0.f,0.f,0.f,0.f,0.f,0.f,0.f};
  v8f s2 = {0.f,0.f,0.f,0.f,0.f,0.f,0.f,0.f};

  for (int i = 0; i < TPW; ++i) {
    const int col0 = (t0 + i) * 16;

    // Prefetch next tile into registers while current tile is consumed
    v4f r0 = {0.f,0.f,0.f,0.f}, r1 = {0.f,0.f,0.f,0.f};
    const bool has = (i + 1 < TPW);
    if (has) {
      const float* ps = row_ptr(o1, o2, (t0 + i + 1) * 16) + srow * D_ + scol;
      r0 = *(const v4f*)(ps);
      r1 = *(const v4f*)(ps + 4);
    }

    __syncthreads();                               // buf[i&1] stores visible

    // B tile from LDS in WMMA layout (conflict-free with LSTRIDE=132)
    const float* bbase = &lds[i & 1][lr * LSTRIDE + koff];
    v2f b[KSTEPS];
#pragma unroll
    for (int k = 0; k < KSTEPS; ++k) b[k] = *(const v2f*)(bbase + 4 * k);

    v8f d0 = {0.f,0.f,0.f,0.f,0.f,0.f,0.f,0.f};
    v8f d1 = {0.f,0.f,0.f,0.f,0.f,0.f,0.f,0.f};
#pragma unroll
    for (int k = 0; k < KSTEPS; k += 2) {          // two chains for ILP
      d0 = wmma_f32(a[k],     b[k],     d0);
      d1 = wmma_f32(a[k + 1], b[k + 1], d1);
    }

    // mask self + positive pair: tiles with row0%B == col0%B, on the tile diagonal.
    // C/D layout: VGPR j, lanes 0-15 -> (M=j, N=lane); lanes 16-31 -> (M=j+8, N=lane-16)
    const bool diag = ((row0 & (B_ - 1)) == (col0 & (B_ - 1)));
#pragma unroll
    for (int j = 0; j < 8; ++j) {
      float e = __expf((d0[j] + d1[j]) * INV_TEMP);
      if (diag && (lane == j || lane == j + 24)) e = 0.0f;
      s1[j] += e;
      s2[j] += e * e;
    }

    if (has) {                                     // fill the other buffer
      float* pd = &lds[(i + 1) & 1][srow * LSTRIDE + scol];
      *(v4f*)(pd)     = r0;
      *(v4f*)(pd + 4) = r1;
    }
  }

  // reduce the 16 N-lanes of each half-wave (xor <16 never crosses halves),
  // each wave owns distinct rows -> direct store, fully deterministic
  const int base = blockIdx.y * N2_ + row0;
#pragma unroll
  for (int j = 0; j < 8; ++j) {
    float v1 = s1[j], v2 = s2[j];
#pragma unroll
    for (int m = 1; m < 16; m <<= 1) {
      v1 += __shfl_xor(v1, m, 32);
      v2 += __shfl_xor(v2, m, 32);
    }
    if (lane == 0)  { s1p[base + j]     = v1; s2p[base + j]     = v2; }  // M = j
    if (lane == 16) { s1p[base + j + 8] = v1; s2p[base + j + 8] = v2; }  // M = j+8
  }
}

// loss = mean_r -log(pos/(pos+Ng)), Ng = max((-tau*N*pos + N*s2/s1)/(1-tau), N*e^{-1/T})
__global__ __launch_bounds__(256)
void hardcl_loss(const float* __restrict__ s1p, const float* __restrict__ s2p,
                 const float* __restrict__ pos, float* __restrict__ out) {
  __shared__ float sh[256];
  const float N      = 2.0f * B_ - 2.0f;           // 8190
  const float clampv = N * expf(-INV_TEMP);        // N * e^{-1/T}
  float acc = 0.0f;
  for (int r = threadIdx.x; r < N2_; r += 256) {
    float a1 = 0.f, a2 = 0.f;
#pragma unroll
    for (int c = 0; c < CS; ++c) {                 // fixed-order partial combine
      a1 += s1p[c * N2_ + r];
      a2 += s2p[c * N2_ + r];
    }
    const float p  = pos[r & (B_ - 1)];
    const float rw = N * a2 / a1;
    float Ng = (-TAU_PLUS * N * p + rw) / (1.0f - TAU_PLUS);
    Ng = fmaxf(Ng, clampv);
    acc += -logf(p / (p + Ng));
  }
  sh[threadIdx.x] = acc;
  __syncthreads();
  for (int s = 128; s > 0; s >>= 1) {
    if (threadIdx.x < s) sh[threadIdx.x] += sh[threadIdx.x + s];
    __syncthreads();
  }
  if (threadIdx.x == 0) out[0] = sh[0] / (float)N2_;
}

extern "C" void kernel_launch(void* const* d_in, const int* in_sizes, int n_in,
                              void* d_out, int out_size, void* d_ws, size_t ws_size,
                              hipStream_t stream) {
  (void)in_sizes; (void)n_in; (void)out_size; (void)ws_size;
  const float* o1 = (const float*)d_in[0];
  const float* o2 = (const float*)d_in[1];
  float* ws  = (float*)d_ws;
  float* s1p = ws;                        // [CS][8192]
  float* s2p = ws + CS * N2_;             // [CS][8192]
  float* pos = ws + 2 * CS * N2_;         // [4096]

  hardcl_pos<<<(B_ * 32) / 256, 256, 0, stream>>>(o1, o2, pos);

  dim3 grid(N2_ / 128, CS);               // 64 row groups x 4 column splits
  hardcl_rowsums<<<grid, 256, 0, stream>>>(o1, o2, s1p, s2p);

  hardcl_loss<<<1, 256, 0, stream>>>(s1p, s2p, pos, (float*)d_out);
}